// Conv4d_49898930045061
// MI455X (gfx1250) — compile-verified
//
#include <hip/hip_runtime.h>

// CDNA5 (gfx1250) implicit-GEMM 4D convolution using V_WMMA_F32_16X16X4_F32.
//
// GEMM view:  D[Cout=32][N=B*oT*oD*oH*oW=352800] = W[32][K=1296] * Xcol[1296][N]
//   K index   k = ci*81 + (((kt*3+kd)*3+kh)*3+kw)
// One wave computes a 32x32 tile of D: two 16x16 M-tiles x two N-tiles, so each
// A fragment (ds_load_2addr_b64) and each offset-table read feeds 4 WMMAs.
//
// Addressing: x-patch offsets are precomputed in BYTES in an LDS table; each
// B element address is  x(SGPR base) + zext32(lane_base + tab[k])  so the
// backend emits the SADDR+voffset global-load form (1 VALU add per load).
//
// Register budget: __launch_bounds__(256, 4) -> up to 256 VGPRs/wave so the
// scheduler can keep all 8 B addresses of a step-pair live and clause the
// global loads (avoids the single-temp s_wait_xcnt serialization).

typedef __attribute__((ext_vector_type(2))) float v2f;
typedef __attribute__((ext_vector_type(8))) float v8f;
typedef __attribute__((ext_vector_type(2))) int   v2i;

// ---- problem constants (fixed by the reference) ----
#define CIN   16
#define T_    16
#define D_    16
#define H_    32
#define W_    32
#define COUT  32
#define OT    14
#define OD    14
#define OH    30
#define OW    30

#define R81   81                  // 3*3*3*3
#define KTOT  (CIN * R81)         // 1296
#define NSP   (OT * OD * OH * OW) // 176400 (per batch)
#define NTOT  (2 * NSP)           // 352800

// x strides in floats: [B][Ci][T][D][H][W]
#define XS_H  W_                  // 32
#define XS_D  (H_ * W_)           // 1024
#define XS_T  (D_ * H_ * W_)      // 16384
#define XS_C  (T_ * D_ * H_ * W_) // 262144
#define XS_B  (CIN * XS_C)        // 4194304

// out strides in floats: [B][Cout][oT][oD][oH][oW]
#define OS_C  NSP                 // 176400
#define OS_B  (COUT * NSP)        // 5644800

// ---- tiling ----
#define BK      48                // K-panel (1296 = 27 * 48), multiple of 4
#define NPANEL  (KTOT / BK)       // 27
#define WAVES   8                 // waves per block (wave32 -> 256 threads)
#define NTILE   32                // N columns per wave (2 x 16)
#define PANEL_ELTS (BK * COUT)    // 1536 floats per panel
#define ELTS_PER_THREAD (PANEL_ELTS / 256) // 6

__global__ __launch_bounds__(256, 4) void conv4d_wmma_f32(
    const float* __restrict__ x,
    const float* __restrict__ w,
    const float* __restrict__ bias,
    float* __restrict__ out)
{
    // Weight panels in a pair-swizzled layout: for k-pair p = k>>1,
    //   ldsA[buf][p*64 + m*2 + (k&1)] = W[m][k]
    // so a lane's A fragment {W[m][k], W[m][k+1]} is one aligned ds_load_b64
    // (and both M-tiles come from one ds_load_2addr_b64).
    __shared__ float ldsA[2][PANEL_ELTS];     // 2 * 6 KB
    __shared__ int   tab[KTOT];               // per-K x offsets in BYTES, 5.2 KB

    const int tid  = threadIdx.x;
    const int lane = tid & 31;
    const int wave = tid >> 5;
    const int row  = lane & 15;               // M row within tile / N column
    const int hi2  = (lane >> 4) << 1;        // 0 for lanes 0-15, 2 for 16-31

    // ---- build per-K x-offset table in bytes (includes ci stride) ----
    for (int i = tid; i < KTOT; i += 256) {
        int ci = i / R81;
        int r  = i - ci * R81;
        int kw = r % 3;
        int kh = (r / 3) % 3;
        int kd = (r / 9) % 3;
        int kt = r / 27;
        tab[i] = 4 * (ci * XS_C + kt * XS_T + kd * XS_D + kh * XS_H + kw);
    }

    // ---- this wave's N tile (two 16-column sub-tiles) ----
    int n0 = (blockIdx.x * WAVES + wave) * NTILE;
    const bool active = (n0 < NTOT);
    if (!active) n0 = 0;                      // clamp: keep addresses legal, skip stores

    // decompose a column index -> (b, nn) and x byte base of its patch origin
    int bA, nnA, bB, nnB;
    unsigned int xb4A, xb4B;
    {
        const int ncol = n0 + row;            // N sub-tile 0
        bA = ncol / NSP;
        nnA = ncol - bA * NSP;
        const int ot = nnA / (OD * OH * OW);
        const int r2 = nnA - ot * (OD * OH * OW);
        const int od = r2 / (OH * OW);
        const int r3 = r2 - od * (OH * OW);
        const int oh = r3 / OW;
        const int ow = r3 - oh * OW;
        xb4A = 4u * (unsigned)(bA * XS_B + ot * XS_T + od * XS_D + oh * XS_H + ow);
    }
    {
        const int ncol = n0 + 16 + row;       // N sub-tile 1
        bB = ncol / NSP;
        nnB = ncol - bB * NSP;
        const int ot = nnB / (OD * OH * OW);
        const int r2 = nnB - ot * (OD * OH * OW);
        const int od = r2 / (OH * OW);
        const int r3 = r2 - od * (OH * OW);
        const int oh = r3 / OW;
        const int ow = r3 - oh * OW;
        xb4B = 4u * (unsigned)(bB * XS_B + ot * XS_T + od * XS_D + oh * XS_H + ow);
    }
    const char* xbyte = (const char*)x;       // uniform SGPR base

    // ---- panel staging helpers (global->regs, regs->LDS) ----
    float rg[ELTS_PER_THREAD];

    auto gload_panel = [&](int p) {
        const int kb = p * BK;
        #pragma unroll
        for (int j = 0; j < ELTS_PER_THREAD; ++j) {
            int idx = tid + j * 256;          // 0 .. 1535
            int m   = idx / BK;
            int kk  = idx - m * BK;
            rg[j] = w[m * KTOT + kb + kk];    // coalesced along kk
        }
    };
    auto store_panel = [&](int bufsel) {
        float* buf = ldsA[bufsel];
        #pragma unroll
        for (int j = 0; j < ELTS_PER_THREAD; ++j) {
            int idx = tid + j * 256;
            int m   = idx / BK;
            int kk  = idx - m * BK;
            buf[(kk >> 1) * (2 * COUT) + m * 2 + (kk & 1)] = rg[j];
        }
    };

    v8f acc0 = {};   // M-tile 0 x N-tile 0
    v8f acc1 = {};   // M-tile 1 x N-tile 0
    v8f acc2 = {};   // M-tile 0 x N-tile 1
    v8f acc3 = {};   // M-tile 1 x N-tile 1

    gload_panel(0);
    store_panel(0);
    __syncthreads();

    for (int p = 0; p < NPANEL; ++p) {
        if (p + 1 < NPANEL) gload_panel(p + 1);   // prefetch next panel into regs

        const float* buf = ldsA[p & 1];
        const int    kb  = p * BK;

        #pragma unroll
        for (int s = 0; s < BK / 4; ++s) {        // 12 K-steps of 4
            const int kl = 4 * s + hi2;           // this half-wave's local k

            // byte offsets for k, k+1 (even index -> 8B-aligned ds_load_b64)
            const v2i o = *(const v2i*)&tab[kb + kl];

            v2f bfA, bfB;
            bfA.x = *(const float*)(xbyte + (unsigned int)(xb4A + (unsigned)o.x));
            bfA.y = *(const float*)(xbyte + (unsigned int)(xb4A + (unsigned)o.y));
            bfB.x = *(const float*)(xbyte + (unsigned int)(xb4B + (unsigned)o.x));
            bfB.y = *(const float*)(xbyte + (unsigned int)(xb4B + (unsigned)o.y));

            const float* ap = &buf[(kl >> 1) * (2 * COUT) + row * 2];
            const v2f a0 = *(const v2f*)ap;            // M-tile 0 (rows 0-15)
            const v2f a1 = *(const v2f*)(ap + 2 * 16); // M-tile 1 (rows 16-31)

            acc0 = __builtin_amdgcn_wmma_f32_16x16x4_f32(
                false, a0, false, bfA, (short)0, acc0, false, false);
            acc1 = __builtin_amdgcn_wmma_f32_16x16x4_f32(
                false, a1, false, bfA, (short)0, acc1, false, false);
            acc2 = __builtin_amdgcn_wmma_f32_16x16x4_f32(
                false, a0, false, bfB, (short)0, acc2, false, false);
            acc3 = __builtin_amdgcn_wmma_f32_16x16x4_f32(
                false, a1, false, bfB, (short)0, acc3, false, false);
        }

        // Write next panel. Safe with a single barrier per iteration: buffer
        // (p+1)&1 was last read in compute(p-1), ordered by the barrier at the
        // end of iteration p-1; compute(p) reads the other buffer.
        if (p + 1 < NPANEL) store_panel((p + 1) & 1);
        __syncthreads();
    }

    // ---- epilogue: bias add + store (C/D layout: VGPR r -> M=r / r+8 per half) ----
    if (active) {
        const int rbase = (lane >> 4) << 3;       // 0 or 8
        float* opA = out + bA * OS_B + nnA;       // column base, N sub-tile 0
        float* opB = out + bB * OS_B + nnB;       // column base, N sub-tile 1
        #pragma unroll
        for (int r = 0; r < 8; ++r) {
            const int m0 = r + rbase;             // row in M-tile 0
            const float bz0 = bias[m0];
            const float bz1 = bias[m0 + 16];
            opA[m0 * OS_C]        = acc0[r] + bz0;
            opA[(m0 + 16) * OS_C] = acc1[r] + bz1;
            opB[m0 * OS_C]        = acc2[r] + bz0;
            opB[(m0 + 16) * OS_C] = acc3[r] + bz1;
        }
    }
}

extern "C" void kernel_launch(void* const* d_in, const int* in_sizes, int n_in,
                              void* d_out, int out_size, void* d_ws, size_t ws_size,
                              hipStream_t stream) {
    const float* x    = (const float*)d_in[0];
    const float* w    = (const float*)d_in[1];
    const float* bias = (const float*)d_in[2];
    float* out        = (float*)d_out;

    const int ntiles = NTOT / NTILE;                  // 11025 wave tiles
    const int blocks = (ntiles + WAVES - 1) / WAVES;  // 1379
    hipLaunchKernelGGL(conv4d_wmma_f32, dim3(blocks), dim3(256), 0, stream,
                       x, w, bias, out);
}